// MeanMessageAggregator_80101140070793
// MI455X (gfx1250) — compile-verified
//
#include <hip/hip_runtime.h>
#include <stdint.h>

// ---------------------------------------------------------------------------
// MeanMessageAggregator for MI455X (gfx1250)
//   means[v]   = segment_mean(messages, node_ids)
//   last_ts[v] = timestamps[segment_max(arange(N), node_ids)]  (0 if absent)
//   present[v] = count > 0
//
// Memory-bound (AI ~ 0.25 FLOP/B). Strategy: TDM (tensor_load_to_lds) streams
// 48-row message tiles into LDS (double buffered, TENSORcnt-synced) so wave
// VMEM issue slots are spent on the f32 scatter atomics, which land in the
// 192MB L2 (accumulator is only ~51MB).
// ---------------------------------------------------------------------------

#define MSG_DIM        128
#define TILE_ROWS      48
#define TILE_ELEMS     (TILE_ROWS * MSG_DIM)
#define TILE_BYTES     (TILE_ELEMS * 4)
#define SCATTER_BLOCK  256
#define WAVES_PER_BLK  (SCATTER_BLOCK / 32)

typedef __attribute__((ext_vector_type(4))) unsigned int v4u_t;
typedef __attribute__((ext_vector_type(4))) int          v4i_t;
typedef __attribute__((ext_vector_type(8))) int          v8i_t;

// Issue one TDM load of a [tile_rows x 128] f32 tile (row-major, stride 128)
// into LDS at byte offset lds_byte_addr. rows_in_bounds clamps the tensor's
// dim1 so tail tiles read OOB rows as zero instead of faulting.
__device__ __forceinline__ void tdm_load_rows_to_lds(const float* gsrc,
                                                     uint32_t lds_byte_addr,
                                                     uint32_t rows_in_bounds,
                                                     uint32_t tile_rows) {
  uint64_t ga = (uint64_t)(uintptr_t)gsrc;

  // D# group 0: count=1 (valid user descriptor), lds_addr, 57-bit global addr,
  // type=2 ("image") in bits [127:126].
  v4u_t g0;
  g0.x = 1u;
  g0.y = lds_byte_addr;
  g0.z = (uint32_t)ga;
  g0.w = ((uint32_t)(ga >> 32) & 0x01FFFFFFu) | (2u << 30);

  // D# group 1: data_size=2 (4B), tensor_dim0=128, tensor_dim1=rows_in_bounds,
  // tile_dim0=128, tile_dim1=tile_rows, tensor_dim0_stride=128. No padding,
  // no multicast, no atomic-barrier, no iterate.
  v8i_t g1;
  g1[0] = (int)(2u << 16);                                              // data_size=4B
  g1[1] = (int)(((uint32_t)MSG_DIM & 0xFFFFu) << 16);                   // dim0 lo16
  g1[2] = (int)((((uint32_t)MSG_DIM >> 16) & 0xFFFFu)                   // dim0 hi16
                | ((rows_in_bounds & 0xFFFFu) << 16));                  // dim1 lo16
  g1[3] = (int)(((rows_in_bounds >> 16) & 0xFFFFu)                      // dim1 hi16
                | ((uint32_t)MSG_DIM << 16));                           // tile_dim0
  g1[4] = (int)(tile_rows & 0xFFFFu);                                   // tile_dim1 (tile_dim2=0)
  g1[5] = (int)MSG_DIM;                                                 // dim0_stride lo32
  g1[6] = 0;
  g1[7] = 0;

  v4i_t z4 = {0, 0, 0, 0};
#if defined(__clang_major__) && (__clang_major__ >= 23)
  v8i_t z8 = {0, 0, 0, 0, 0, 0, 0, 0};
  __builtin_amdgcn_tensor_load_to_lds(g0, g1, z4, z4, z8, 0);
#else
  __builtin_amdgcn_tensor_load_to_lds(g0, g1, z4, z4, 0);
#endif
}

// ---------------------------------------------------------------------------
// Kernel 1: initialize accumulators (means region of d_out, counts, last_idx)
// ---------------------------------------------------------------------------
__global__ void mma_init_kernel(float* __restrict__ means,
                                unsigned int* __restrict__ counts,
                                int* __restrict__ last_idx,
                                int num_nodes) {
  const int stride = (int)(gridDim.x * blockDim.x);
  const int tid = (int)(blockIdx.x * blockDim.x + threadIdx.x);
  const long long total = (long long)num_nodes * MSG_DIM;
  for (long long i = tid; i < total; i += stride) means[i] = 0.0f;
  for (int i = tid; i < num_nodes; i += stride) {
    counts[i] = 0u;
    last_idx[i] = (int)0x80000000;  // INT_MIN, matches segment_max identity
  }
}

// ---------------------------------------------------------------------------
// Kernel 2: TDM-staged tile scatter.
//   wave 0 double-buffers 48x128 f32 tiles into LDS via tensor_load_to_lds;
//   every 32-lane group consumes one row: 4 f32 scatter-atomics per lane,
//   lane 0 bumps count and the running max of the message index.
// ---------------------------------------------------------------------------
__global__ __launch_bounds__(SCATTER_BLOCK)
void mma_scatter_kernel(const float* __restrict__ messages,
                        const int* __restrict__ node_ids,
                        float* __restrict__ sums,
                        unsigned int* __restrict__ counts,
                        int* __restrict__ last_idx,
                        int n_msg) {
  __shared__ __align__(16) float tiles[2][TILE_ELEMS];

  const int wave = (int)(threadIdx.x >> 5);
  const int lane = (int)(threadIdx.x & 31);
  const int n_tiles = (n_msg + TILE_ROWS - 1) / TILE_ROWS;
  const int t0 = (int)blockIdx.x;
  const int tstride = (int)gridDim.x;
  if (t0 >= n_tiles) return;  // uniform per block

  // Flat shared pointers: low 32 bits are the LDS byte offset.
  const uint32_t lds_buf[2] = {
      (uint32_t)(uintptr_t)&tiles[0][0],
      (uint32_t)(uintptr_t)&tiles[1][0],
  };

  // Prime the pipeline: TDM tile t0 -> buffer 0.
  if (wave == 0) {
    uint32_t r0 = (uint32_t)t0 * TILE_ROWS;
    tdm_load_rows_to_lds(messages + (size_t)r0 * MSG_DIM, lds_buf[0],
                         (uint32_t)n_msg - r0, TILE_ROWS);
  }

  int buf = 0;
  for (int t = t0; t < n_tiles; t += tstride) {
    const int tn = t + tstride;
    if (wave == 0) {
      if (tn < n_tiles) {
        uint32_t r0n = (uint32_t)tn * TILE_ROWS;
        // Warm L2 for the next tile's node ids (global_prefetch_b8).
        __builtin_prefetch(node_ids + r0n, 0, 1);
        tdm_load_rows_to_lds(messages + (size_t)r0n * MSG_DIM,
                             lds_buf[buf ^ 1], (uint32_t)n_msg - r0n,
                             TILE_ROWS);
        // Two TDM ops in flight, in-order: <=1 means tile t has landed.
        __builtin_amdgcn_s_wait_tensorcnt((short)1);
      } else {
        __builtin_amdgcn_s_wait_tensorcnt((short)0);
      }
    }
    __syncthreads();  // publish TDM-written LDS to all waves

    const int r0 = t * TILE_ROWS;
    const int rows = (n_msg - r0 < TILE_ROWS) ? (n_msg - r0) : TILE_ROWS;
    for (int row = wave; row < rows; row += WAVES_PER_BLK) {
      const int grow = r0 + row;
      const int node = node_ids[grow];  // wave-uniform
      const float4 v =
          *(const float4*)&tiles[buf][row * MSG_DIM + lane * 4];
      float* dst = sums + (size_t)node * MSG_DIM + lane * 4;
      atomicAdd(dst + 0, v.x);
      atomicAdd(dst + 1, v.y);
      atomicAdd(dst + 2, v.z);
      atomicAdd(dst + 3, v.w);
      if (lane == 0) {
        atomicAdd(&counts[node], 1u);
        atomicMax(&last_idx[node], grow);
      }
    }
    __syncthreads();  // all reads of this buffer done before TDM reuses it
    buf ^= 1;
  }
}

// ---------------------------------------------------------------------------
// Kernel 3: finalize. One node per 32-lane group: divide sums in place,
// gather last timestamp, emit presence flag.
// ---------------------------------------------------------------------------
__global__ __launch_bounds__(256)
void mma_finalize_kernel(float* __restrict__ means,
                         const unsigned int* __restrict__ counts,
                         const int* __restrict__ last_idx,
                         const float* __restrict__ timestamps,
                         float* __restrict__ last_ts,
                         float* __restrict__ present,
                         int num_nodes) {
  const int node = (int)((blockIdx.x * blockDim.x + threadIdx.x) >> 5);
  const int lane = (int)(threadIdx.x & 31);
  if (node >= num_nodes) return;

  const unsigned int c = counts[node];
  const float denom = (float)(c > 0u ? c : 1u);

  float4* p = (float4*)(means + (size_t)node * MSG_DIM + lane * 4);
  float4 x = *p;
  x.x /= denom; x.y /= denom; x.z /= denom; x.w /= denom;
  *p = x;

  if (lane == 0) {
    const int li = last_idx[node];
    const bool has = (c > 0u);
    present[node] = has ? 1.0f : 0.0f;
    last_ts[node] = has ? timestamps[li] : 0.0f;
  }
}

// ---------------------------------------------------------------------------
extern "C" void kernel_launch(void* const* d_in, const int* in_sizes, int n_in,
                              void* d_out, int out_size, void* d_ws,
                              size_t ws_size, hipStream_t stream) {
  (void)n_in; (void)ws_size;

  const float* messages   = (const float*)d_in[0];
  const float* timestamps = (const float*)d_in[1];
  const int*   node_ids   = (const int*)d_in[2];

  const int n_msg = in_sizes[1];                    // #messages (= len(timestamps))
  const int num_nodes = out_size / (MSG_DIM + 2);   // means + last_ts + present

  // Output layout: [means V*128][last_ts V][present V]
  float* means   = (float*)d_out;
  float* last_ts = means + (size_t)num_nodes * MSG_DIM;
  float* present = last_ts + num_nodes;

  // Workspace: counts (u32[V]) then last_idx (i32[V]).
  unsigned int* counts = (unsigned int*)d_ws;
  int* last_idx = (int*)(counts + num_nodes);

  // 1) init accumulators
  {
    const int threads = 256;
    int blocks = (int)(((long long)num_nodes * MSG_DIM + threads - 1) / threads);
    if (blocks > 4096) blocks = 4096;
    mma_init_kernel<<<blocks, threads, 0, stream>>>(means, counts, last_idx,
                                                    num_nodes);
  }

  // 2) TDM-staged atomic scatter
  {
    const int n_tiles = (n_msg + TILE_ROWS - 1) / TILE_ROWS;
    int blocks = n_tiles < 4096 ? n_tiles : 4096;
    if (blocks < 1) blocks = 1;
    mma_scatter_kernel<<<blocks, SCATTER_BLOCK, 0, stream>>>(
        messages, node_ids, means, counts, last_idx, n_msg);
  }

  // 3) finalize (8 nodes per 256-thread block)
  {
    const int nodes_per_block = 256 / 32;
    const int blocks = (num_nodes + nodes_per_block - 1) / nodes_per_block;
    mma_finalize_kernel<<<blocks, 256, 0, stream>>>(
        means, counts, last_idx, timestamps, last_ts, present, num_nodes);
  }
}